// context_window_43233140801616
// MI455X (gfx1250) — compile-verified
//
#include <hip/hip_runtime.h>

// Context-window (grouped identity conv1d) for x: (32, 80, 3000) fp32,
// out: (32, 80*context_len, 3000) fp32.  Pure data movement:
//   read 30.7MB once, write 338MB once  ->  ~16us roofline @ 23.3 TB/s.
// Strategy: stage each input row-tile (+halo) into LDS once via CDNA5
// async-to-LDS loads (ASYNCcnt path), then emit the context_len shifted
// copies with 128-bit non-temporal global stores (store-bound; keep L2 clean).

#define TBLOCK 256
#define T_DIM 3000
#define T_TILE 1000
#define MAXP 16
#define NTILES ((T_DIM + T_TILE - 1) / T_TILE)
// capacity: tile + max halo + one extra block-stride of slack for the
// full-EXEC padded async staging loop
#define LDS_CAP (T_TILE + 2 * MAXP + TBLOCK)

typedef __attribute__((ext_vector_type(4))) float v4f;
typedef __attribute__((address_space(3))) int lds_int_t;  // LDS (shared)
typedef __attribute__((address_space(1))) int g_int_t;    // global (__device__)

__global__ __launch_bounds__(TBLOCK) void context_window_kernel(
    const float* __restrict__ x, const int* __restrict__ pl,
    const int* __restrict__ pr, float* __restrict__ out, int rows) {
  __shared__ float tile[LDS_CAP];

  const int l = pl[0];
  const int r = pr[0];
  int P = (l > r) ? l : r;
  if (P > MAXP) P = MAXP;                  // LDS capacity guard
  const int lag = (r > l) ? (r - l) : 0;
  const int CL = l + r + 1;                // context_len

  const int bid = blockIdx.x;
  const int row = bid / NTILES;            // row = b*F + f
  const int tl = bid - row * NTILES;
  if (row >= rows) return;
  const int t0 = tl * T_TILE;
  const int tid = threadIdx.x;

  const int halo_n = T_TILE + 2 * P;       // valid staged span
  const long rowBase = (long)row * T_DIM;

  // ---- Stage x[row, t0-P .. t0+T_TILE+P) into LDS, zeros outside [0,T) ----
#if __has_builtin(__builtin_amdgcn_global_load_async_to_lds_b32)
  {
    // Pad iteration count so EXEC is all-ones for every async issue; clamp
    // global addresses into range (garbage slots fixed up / unused below).
    const int halo_pad = (halo_n + TBLOCK - 1) & ~(TBLOCK - 1);
    for (int i = tid; i < halo_pad; i += TBLOCK) {
      int t = t0 - P + i;
      int tc = t < 0 ? 0 : (t >= T_DIM ? T_DIM - 1 : t);
      __builtin_amdgcn_global_load_async_to_lds_b32(
          (g_int_t*)(x + rowBase + tc),
          (lds_int_t*)(tile + i), /*offset=*/0, /*cpol=*/0);
    }
#if __has_builtin(__builtin_amdgcn_s_wait_asynccnt)
    __builtin_amdgcn_s_wait_asynccnt(0);
#else
    asm volatile("s_wait_asynccnt 0x0" ::: "memory");
#endif
    // Each thread fixes exactly the slots it loaded (its ASYNCcnt is 0 now).
    for (int i = tid; i < halo_n; i += TBLOCK) {
      int t = t0 - P + i;
      if (t < 0 || t >= T_DIM) tile[i] = 0.0f;
    }
  }
#else
  for (int i = tid; i < halo_n; i += TBLOCK) {
    int t = t0 - P + i;
    tile[i] = (t >= 0 && t < T_DIM) ? x[rowBase + t] : 0.0f;
  }
#endif
  __syncthreads();

  // ---- Emit CL shifted copies: out[(row*CL + c)*T + t0 + j] = tile[j+c+lag]
  const int tileLen = (t0 + T_TILE <= T_DIM) ? T_TILE : (T_DIM - t0);
  const int nvec = tileLen >> 2;           // 16B-aligned float4 stores
  for (int c = 0; c < CL; ++c) {
    const long base = ((long)row * CL + c) * T_DIM + t0;
    const int off = c + lag;               // LDS shift, 0..2P
    for (int j4 = tid; j4 < nvec; j4 += TBLOCK) {
      const int j = j4 << 2;
      v4f v = {tile[j + off], tile[j + off + 1], tile[j + off + 2],
               tile[j + off + 3]};
      __builtin_nontemporal_store(v, (v4f*)(out + base + (long)j));
    }
    for (int j = (nvec << 2) + tid; j < tileLen; j += TBLOCK) {
      out[base + j] = tile[j + off];       // scalar tail (none for T=3000)
    }
  }
}

extern "C" void kernel_launch(void* const* d_in, const int* in_sizes, int n_in,
                              void* d_out, int out_size, void* d_ws,
                              size_t ws_size, hipStream_t stream) {
  const float* x = (const float*)d_in[0];
  const int* pl = (const int*)d_in[1];   // left_frames (device scalar)
  const int* pr = (const int*)d_in[2];   // right_frames (device scalar)
  float* out = (float*)d_out;

  const int rows = in_sizes[0] / T_DIM;  // B*F = 2560
  dim3 grid(rows * NTILES);
  dim3 block(TBLOCK);
  hipLaunchKernelGGL(context_window_kernel, grid, block, 0, stream, x, pl, pr,
                     out, rows);
}